// ElevationLoss_59889023975542
// MI455X (gfx1250) — compile-verified
//
#include <hip/hip_runtime.h>

typedef __attribute__((ext_vector_type(2))) float v2f;
typedef __attribute__((ext_vector_type(8))) float v8f;

static constexpr int kH = 1024;
static constexpr int kW = 1024;
static constexpr int kThreads = 256;          // 8 waves (wave32)
static constexpr int kWaves   = kThreads / 32;
static constexpr int kRow     = kW + 8;       // 4-float left pad keeps b128 dests 16B aligned

// Exact, deterministic 32-lane sum using V_WMMA_F32_16X16X4_F32.
// A (16x4): lane L<16 supplies A[L][0]=v, A[L][1]=0; lane L>=16 supplies
// A[L-16][2]=v, A[L-16][3]=0.  B = all-ones (layout-independent), C = 0.
// => D[m][n] = v[m] + v[m+16] for every n.  Lane sum of d[0..7] gives the
// half-sum (m=0..7 on low lanes, m=8..15 on high); xor-16 + add completes it.
__device__ __forceinline__ float wave_sum_wmma(float v) {
    v2f a; a[0] = v;    a[1] = 0.0f;
    v2f b; b[0] = 1.0f; b[1] = 1.0f;
    v8f c = {};
    v8f d = __builtin_amdgcn_wmma_f32_16x16x4_f32(
        /*neg_a=*/false, a, /*neg_b=*/false, b,
        /*c_mod=*/(short)0, c, /*reuse_a=*/false, /*reuse_b=*/false);
    float h = ((d[0] + d[1]) + (d[2] + d[3])) + ((d[4] + d[5]) + (d[6] + d[7]));
    h += __shfl_xor(h, 16, 32);
    return h;
}

// Async copy 16 bytes/lane from global (SGPR base + VGPR byte offset) into LDS.
// GLOBAL_LOAD_ASYNC_TO_LDS_B128, tracked on ASYNCcnt (CDNA5 ISA).
__device__ __forceinline__ void async_ld_b128(unsigned lds_off, unsigned long long base,
                                              unsigned byte_off) {
    asm volatile("global_load_async_to_lds_b128 %0, %1, %2"
                 :: "v"(lds_off), "v"(byte_off), "s"(base)
                 : "memory");
}

__device__ __forceinline__ void wait_asynccnt0() {
    asm volatile("s_wait_asynccnt 0" ::: "memory");
}

// One block per (batch, row). 256 threads x 4 pixels = 1024 = W.
__global__ __launch_bounds__(kThreads)
void elev_loss_rows(const float* __restrict__ pred,
                    const float* __restrict__ hgt,
                    const float* __restrict__ gt,
                    float* __restrict__ partial,
                    int H, int W)
{
    __shared__ float s_gt[3][kRow];     // data at [4 + x], pads at [3] and [4 + kW]
    __shared__ float s_h [3][kRow];
    __shared__ float s_wave[kWaves];

    const int row = blockIdx.x;         // 0 .. B*H-1
    const int b   = row / H;
    const int y   = row - b * H;
    const int t   = threadIdx.x;
    const int x0  = t * 4;
    const size_t HW = (size_t)H * (size_t)W;

    const float* gt_b = gt  + (size_t)b * HW;
    const float* h_b  = hgt + (size_t)b * HW;
    const float* p0r  = pred + (size_t)b * 2 * HW + (size_t)y * W;  // channel 0 row
    const float* p1r  = p0r + HW;                                   // channel 1 row

    // Kick off pred loads first so they fly during LDS staging + barrier.
    const float4 q0 = *(const float4*)(p0r + x0);
    const float4 q1 = *(const float4*)(p1r + x0);

    // Stage 3 rows of gt and heights via async global->LDS b128 (zero OOB rows).
    const unsigned boff = (unsigned)(x0 * sizeof(float));
    #pragma unroll
    for (int r = 0; r < 3; ++r) {
        const int yy = y + r - 1;
        const unsigned lgt = (unsigned)(uintptr_t)&s_gt[r][4 + x0];
        const unsigned lh  = (unsigned)(uintptr_t)&s_h [r][4 + x0];
        if (yy >= 0 && yy < H) {        // uniform per block: no divergence
            async_ld_b128(lgt, (unsigned long long)(gt_b + (size_t)yy * W), boff);
            async_ld_b128(lh,  (unsigned long long)(h_b  + (size_t)yy * W), boff);
        } else {
            *(float4*)&s_gt[r][4 + x0] = make_float4(0.f, 0.f, 0.f, 0.f);
            *(float4*)&s_h [r][4 + x0] = make_float4(0.f, 0.f, 0.f, 0.f);
        }
    }
    if (t == 0) {
        #pragma unroll
        for (int r = 0; r < 3; ++r) {
            s_gt[r][3] = 0.f; s_gt[r][4 + kW] = 0.f;   // column pads
            s_h [r][3] = 0.f; s_h [r][4 + kW] = 0.f;
        }
    }
    wait_asynccnt0();                   // this wave's async LDS writes landed
    __syncthreads();                    // all waves' writes visible

    // Unified prediction for the 4 centers (argmax tie -> channel 0).
    float u[4];
    u[0] = (q0.x >= q1.x) ? q0.x : -q1.x;
    u[1] = (q0.y >= q1.y) ? q0.y : -q1.y;
    u[2] = (q0.z >= q1.z) ? q0.z : -q1.z;
    u[3] = (q0.w >= q1.w) ? q0.w : -q1.w;

    // 3x6 neighborhood window covering all neighbors of the 4 centers.
    // neighbor x = x0 + k + cc - 1  ->  LDS index 4 + x = x0 + k + cc + 3.
    float gr[3][6], hr[3][6];
    #pragma unroll
    for (int r = 0; r < 3; ++r)
        #pragma unroll
        for (int i = 0; i < 6; ++i) {
            gr[r][i] = s_gt[r][x0 + 3 + i];
            hr[r][i] = s_h [r][x0 + 3 + i];
        }

    // g in {-1,0,1}: masks collapse to keep = (g != 0) && (g*delta <= 0),
    // contribution = keep ? (1 - g*u) : 0   (zero-pad neighbors have g==0).
    float acc = 0.f;
    #pragma unroll
    for (int k = 0; k < 4; ++k) {
        const float uc = u[k];
        const float hc = hr[1][k + 1];          // center height
        #pragma unroll
        for (int r = 0; r < 3; ++r) {
            #pragma unroll
            for (int cc = 0; cc < 3; ++cc) {
                const float g     = gr[r][k + cc];
                const float delta = hc - hr[r][k + cc];
                const float gd    = g * delta;
                const float s     = fmaf(-g, uc, 1.f);       // score = 1 - g*u
                const bool keep   = (g != 0.f) && (gd <= 0.f);
                acc += keep ? s : 0.f;
            }
        }
    }

    // Deterministic block reduction: WMMA wave-sum, then 8 wave values via LDS.
    const float wsum = wave_sum_wmma(acc);
    const int lane = t & 31, wid = t >> 5;
    if (lane == 0) s_wave[wid] = wsum;
    __syncthreads();
    if (t == 0) {
        float s = 0.f;
        #pragma unroll
        for (int i = 0; i < kWaves; ++i) s += s_wave[i];
        partial[row] = s;
    }
}

// Single-block final reduction over B*H partials (fixed order -> deterministic).
__global__ __launch_bounds__(kThreads)
void elev_loss_final(const float* __restrict__ partial,
                     float* __restrict__ out, int n)
{
    __shared__ float s_wave[kWaves];
    const int t = threadIdx.x;
    float acc = 0.f;
    for (int i = t; i < n; i += kThreads)   // n multiple of 256: no divergence
        acc += partial[i];
    const float wsum = wave_sum_wmma(acc);
    const int lane = t & 31, wid = t >> 5;
    if (lane == 0) s_wave[wid] = wsum;
    __syncthreads();
    if (t == 0) {
        float s = 0.f;
        #pragma unroll
        for (int i = 0; i < kWaves; ++i) s += s_wave[i];
        out[0] = s;
    }
}

extern "C" void kernel_launch(void* const* d_in, const int* in_sizes, int n_in,
                              void* d_out, int out_size, void* d_ws, size_t ws_size,
                              hipStream_t stream) {
    const float* pred = (const float*)d_in[0];   // (B, 2, H, W) f32
    const float* hgt  = (const float*)d_in[1];   // (B, 1, H, W) f32
    const float* gt   = (const float*)d_in[2];   // (B, 1, H, W) f32
    float* out     = (float*)d_out;              // 1 float
    float* partial = (float*)d_ws;               // B*H floats of scratch

    const int H = kH, W = kW;
    const int BH = in_sizes[1] / W;              // B * H (= 4096)

    elev_loss_rows <<<BH, kThreads, 0, stream>>>(pred, hgt, gt, partial, H, W);
    elev_loss_final<<<1,  kThreads, 0, stream>>>(partial, out, BH);
}